// ProbabilisticMap_13597866459174
// MI455X (gfx1250) — compile-verified
//
#include <hip/hip_runtime.h>
#include <hip/hip_bf16.h>

// ProbabilisticMap for MI455X (gfx1250, wave32).
// Stage 1: per-batch (T x K)(K x 6) f32 matmul via v_wmma_f32_16x16x4_f32,
//          fusing mean (weights w) and covariance (weights w^2) reductions
//          into one 16x16x16 accumulation per 16-t tile. Bernstein weights
//          now use a binomial LUT + branch-free pow-by-squaring (no divides,
//          no divergent loops).
// Stage 2: bandwidth-bound 128 MB Gaussian-map stream; 4 t-values per thread
//          -> global_store_b128, full 256B line per half-wave per store.

typedef __attribute__((ext_vector_type(2))) float v2f;
typedef __attribute__((ext_vector_type(8))) float v8f;

#define NUM_CP 8
#define BATCH  128
#define TSZ    64
#define MAPW   64
#define MAPH   64

// binom(n,k) for n,k in [0,7]; 0 where k > n (kills invalid terms, matching
// the reference's `valid` mask on the gammaln formulation).
__device__ __constant__ float BINOM[64] = {
    1, 0,  0,  0,  0,  0, 0, 0,
    1, 1,  0,  0,  0,  0, 0, 0,
    1, 2,  1,  0,  0,  0, 0, 0,
    1, 3,  3,  1,  0,  0, 0, 0,
    1, 4,  6,  4,  1,  0, 0, 0,
    1, 5, 10, 10,  5,  1, 0, 0,
    1, 6, 15, 20, 15,  6, 1, 0,
    1, 7, 21, 35, 35, 21, 7, 1,
};

// x^e for 0 <= e <= 7, branch-free pow-by-squaring (3 muls + 3 selects).
__device__ __forceinline__ float powi7(float x, int e) {
    float r = 1.0f;
    float p = x;
    if (e & 1) r *= p;
    p *= p;
    if (e & 2) r *= p;
    p *= p;
    if (e & 4) r *= p;
    return r;
}

// Bernstein basis weight: binom(n,k) * t^k * (1-t)^max(n-k,0).
__device__ __forceinline__ float bern_w(int n, int k, float t) {
    const float coef = BINOM[n * 8 + k];
    const int pot = (n - k) > 0 ? (n - k) : 0;
    return coef * powi7(t, k) * powi7(1.0f - t, pot);
}

// One block per batch; 4 waves, each wave owns a 16-t tile.
// D(16x16) = A(16x16) x B(16x16) accumulated as 4 chained K=4 WMMA steps.
//   A[t, K]   : K in [0,8)  -> w_K(t)        (drives mean)
//               K in [8,16) -> w_{K-8}(t)^2  (drives covariance)
//   B[K, N]   : K<8 : N==0 -> mean_x[k], N==1 -> mean_y[k]
//               K>=8: N==2 -> c00[k], N==3 -> c01[k], N==4 -> c11[k]
__global__ void __launch_bounds__(128)
bezier_params_kernel(const float* __restrict__ cp_means,   // (8,128,2)
                     const int*   __restrict__ num_cps,    // (128,)
                     const float* __restrict__ cp_cov,     // (8,128,2,2)
                     float* __restrict__ params)           // (128,64,6)
{
    __shared__ float tile[4 * 256];

    const int b    = blockIdx.x;
    const int tid  = threadIdx.x;
    const int wave = tid >> 5;
    const int lane = tid & 31;
    const int half = lane >> 4;   // lane group 0/1
    const int mn   = lane & 15;   // M for A-rows, N for B-cols, per ISA layout
    const int t0   = wave * 16;
    const int n    = num_cps[b] - 1;
    const float t  = (float)(t0 + mn) * (1.0f / 63.0f);   // linspace(0,1,64)

    v8f acc = {0.0f, 0.0f, 0.0f, 0.0f, 0.0f, 0.0f, 0.0f, 0.0f};

    #pragma unroll
    for (int ks = 0; ks < 4; ++ks) {
        v2f afrag, bfrag;
        #pragma unroll
        for (int v = 0; v < 2; ++v) {
            // ISA 16x4 f32 layout: VGPR v, lane-half h -> K = 4*ks + 2*h + v
            const int K = ks * 4 + 2 * half + v;
            const int k = K & 7;

            const float w = bern_w(n, k, t);
            const float aval = (K < 8) ? w : w * w;

            float bval = 0.0f;
            if (K < 8) {
                if (mn == 0)      bval = cp_means[(k * BATCH + b) * 2 + 0];
                else if (mn == 1) bval = cp_means[(k * BATCH + b) * 2 + 1];
            } else {
                if (mn == 2)      bval = cp_cov[(k * BATCH + b) * 4 + 0]; // c00
                else if (mn == 3) bval = cp_cov[(k * BATCH + b) * 4 + 1]; // c01
                else if (mn == 4) bval = cp_cov[(k * BATCH + b) * 4 + 3]; // c11
            }
            if (v == 0) { afrag.x = aval; bfrag.x = bval; }
            else        { afrag.y = aval; bfrag.y = bval; }
        }
        // 8 args: (neg_a, A, neg_b, B, c_mod, C, reuse_a, reuse_b)
        acc = __builtin_amdgcn_wmma_f32_16x16x4_f32(
            false, afrag, false, bfrag, (short)0, acc, false, false);
    }

    // D layout: VGPR r, lane -> M = r + 8*half, N = mn
    #pragma unroll
    for (int r = 0; r < 8; ++r)
        tile[wave * 256 + (r + 8 * half) * 16 + mn] = acc[r];

    __syncthreads();

    if (lane < 16) {
        const float* row = &tile[wave * 256 + lane * 16];
        const float mx  = row[0], my  = row[1];
        const float c00 = row[2], c01 = row[3], c11 = row[4];
        const float det  = c00 * c11 - c01 * c01;
        const float rdet = __builtin_amdgcn_rcpf(det);
        // 1 / sqrt((2*pi)^2 * det) = (1/(2*pi)) * rsqrt(det)
        const float pref = 0.15915494309189535f * __builtin_amdgcn_rsqf(det);

        float* o = &params[(b * TSZ + t0 + lane) * 6];
        o[0] = mx;
        o[1] = my;
        o[2] = c11 * rdet;    // inv00
        o[3] = -c01 * rdet;   // inv01 (== inv10)
        o[4] = c00 * rdet;    // inv11
        o[5] = pref;
    }
}

// One block per (b, x): 64 y * 64 t = 4096 outputs, 256 threads.
// Thread -> t group of 4: tq = (tid&15)*4  (16B-aligned, b128 stores),
//           y = (tid>>4) + 16*i, i in [0,4).
__global__ void __launch_bounds__(256)
gauss_map_kernel(const float* __restrict__ params,   // (128,64,6)
                 float* __restrict__ out)            // (128,64,64,64)
{
    __shared__ float sp[TSZ * 6];

    const int b = blockIdx.x >> 6;
    const int x = blockIdx.x & 63;
    const int tid = threadIdx.x;

    for (int i = tid; i < TSZ * 6; i += 256)
        sp[i] = params[b * (TSZ * 6) + i];
    __syncthreads();

    const int tq = (tid & 15) * 4;   // first of 4 consecutive t values
    const int y0 = tid >> 4;         // 0..15

    // exp(-0.5*q)/divisor = pref * exp2(c0 + c1*dy + c2*dy^2), log2(e) folded.
    const float L2E = 1.4426950408889634f;
    float myv[4], c0[4], c1[4], c2[4], pf[4];
    #pragma unroll
    for (int j = 0; j < 4; ++j) {
        const float* p = &sp[(tq + j) * 6];
        const float mx  = p[0];
        myv[j]          = p[1];
        const float i00 = p[2], i01 = p[3], i11 = p[4];
        pf[j]           = p[5];
        const float dx  = (float)x - mx;
        c0[j] = -0.5f * L2E * i00 * dx * dx;
        c1[j] = -L2E * i01 * dx;
        c2[j] = -0.5f * L2E * i11;
    }

    float4* orow = (float4*)(out + (size_t)(b * MAPW + x) * (MAPH * TSZ) + tq);

    #pragma unroll
    for (int y = y0; y < MAPH; y += 16) {
        float4 v;
        float dy;
        dy = (float)y - myv[0];
        v.x = pf[0] * __builtin_amdgcn_exp2f(c0[0] + (c1[0] + c2[0] * dy) * dy);
        dy = (float)y - myv[1];
        v.y = pf[1] * __builtin_amdgcn_exp2f(c0[1] + (c1[1] + c2[1] * dy) * dy);
        dy = (float)y - myv[2];
        v.z = pf[2] * __builtin_amdgcn_exp2f(c0[2] + (c1[2] + c2[2] * dy) * dy);
        dy = (float)y - myv[3];
        v.w = pf[3] * __builtin_amdgcn_exp2f(c0[3] + (c1[3] + c2[3] * dy) * dy);
        orow[(size_t)y * 16] = v;   // == out[..., y, tq..tq+3]
    }
}

extern "C" void kernel_launch(void* const* d_in, const int* in_sizes, int n_in,
                              void* d_out, int out_size, void* d_ws, size_t ws_size,
                              hipStream_t stream) {
    const float* cp_means = (const float*)d_in[0];   // (8,128,2) f32
    const int*   num_cps  = (const int*)d_in[1];     // (128,)    i32
    const float* cp_cov   = (const float*)d_in[2];   // (8,128,2,2) f32
    float* params = (float*)d_ws;                    // 128*64*6 f32 = 192 KB
    float* out    = (float*)d_out;                   // (128,64,64,64) f32

    bezier_params_kernel<<<BATCH, 128, 0, stream>>>(cp_means, num_cps, cp_cov, params);
    gauss_map_kernel<<<BATCH * MAPW, 256, 0, stream>>>(params, out);
}